// TGATLayer_34222299414741
// MI455X (gfx1250) — compile-verified
//
#include <hip/hip_runtime.h>
#include <math.h>

#define DD 128
#define NHEAD 4
#define LRELU 0.2f
#define EPS_LN 1e-5f
#define EPS_SM 1e-16f

typedef __attribute__((ext_vector_type(2))) float v2f;
typedef __attribute__((ext_vector_type(8))) float v8f;

__device__ __forceinline__ float waveReduceSum(float v) {
#pragma unroll
    for (int m = 16; m >= 1; m >>= 1) v += __shfl_xor(v, m, 32);
    return v;
}

__device__ __forceinline__ unsigned mapf(float f) {
    unsigned u = __float_as_uint(f);
    return (u & 0x80000000u) ? ~u : (u | 0x80000000u);
}
__device__ __forceinline__ float unmapf(unsigned u) {
    return (u & 0x80000000u) ? __uint_as_float(u ^ 0x80000000u) : __uint_as_float(~u);
}

// ---------------- LayerNorm 1: one wave per node ----------------
__global__ void ln1_kernel(const float* __restrict__ x, const float* __restrict__ g,
                           const float* __restrict__ b, float* __restrict__ xt, int n) {
    int wid = (blockIdx.x * blockDim.x + threadIdx.x) >> 5;
    int lane = threadIdx.x & 31;
    if (wid >= n) return;
    const float4 xv = reinterpret_cast<const float4*>(x + (size_t)wid * DD)[lane];
    float mean = waveReduceSum(xv.x + xv.y + xv.z + xv.w) * (1.0f / DD);
    float dx = xv.x - mean, dy = xv.y - mean, dz = xv.z - mean, dw = xv.w - mean;
    float var = waveReduceSum(dx * dx + dy * dy + dz * dz + dw * dw) * (1.0f / DD);
    float inv = rsqrtf(var + EPS_LN);
    float4 gv = reinterpret_cast<const float4*>(g)[lane];
    float4 bv = reinterpret_cast<const float4*>(b)[lane];
    float4 o;
    o.x = dx * inv * gv.x + bv.x;
    o.y = dy * inv * gv.y + bv.y;
    o.z = dz * inv * gv.z + bv.z;
    o.w = dw * inv * gv.w + bv.w;
    reinterpret_cast<float4*>(xt + (size_t)wid * DD)[lane] = o;
}

// ---------------- GEMM: h = xt @ W, fp32 WMMA 16x16x4 ----------------
// Block: 256 threads = 8 waves. Block handles 16 rows; wave w handles col tile w (16 cols).
__global__ void gemm_wmma_kernel(const float* __restrict__ A, const float* __restrict__ W,
                                 float* __restrict__ Hout, int n) {
    __shared__ float sA[16 * DD];
    int rowBase = blockIdx.x * 16;
    int tid = threadIdx.x;
    for (int i = tid; i < 16 * DD; i += blockDim.x) {
        int r = rowBase + (i >> 7);
        sA[i] = (r < n) ? A[(size_t)r * DD + (i & 127)] : 0.0f;
    }
    __syncthreads();
    int wave = tid >> 5;
    int lane = tid & 31;
    int colBase = wave * 16;
    int half = lane >> 4;  // 0: K pair {0,1}; 1: K pair {2,3} (ISA 32-bit 16x4 A layout)
    int l15 = lane & 15;
    v8f c = {};
#pragma unroll
    for (int kk = 0; kk < DD; kk += 4) {
        int k0 = kk + half * 2;
        v2f a, bb;
        a[0] = sA[l15 * DD + k0];
        a[1] = sA[l15 * DD + k0 + 1];
        bb[0] = W[(size_t)k0 * DD + colBase + l15];
        bb[1] = W[(size_t)(k0 + 1) * DD + colBase + l15];
        c = __builtin_amdgcn_wmma_f32_16x16x4_f32(false, a, false, bb, (short)0, c, false, false);
    }
    // C/D layout: VGPR i -> M=i (lanes 0-15) / M=8+i (lanes 16-31), N = lane&15
    int mOff = half * 8;
#pragma unroll
    for (int i = 0; i < 8; ++i) {
        int r = rowBase + mOff + i;
        if (r < n) Hout[(size_t)r * DD + colBase + l15] = c[i];
    }
}

// ---------------- Attention scores: one wave per node ----------------
__global__ void score_kernel(const float* __restrict__ Hm, const float* __restrict__ ws_,
                             const float* __restrict__ wd_, float* __restrict__ a_src,
                             float* __restrict__ a_dst, int n) {
    int wid = (blockIdx.x * blockDim.x + threadIdx.x) >> 5;
    int lane = threadIdx.x & 31;
    if (wid >= n) return;
    float4 hv = reinterpret_cast<const float4*>(Hm + (size_t)wid * DD)[lane];
    float4 s4 = reinterpret_cast<const float4*>(ws_)[lane];
    float4 d4 = reinterpret_cast<const float4*>(wd_)[lane];
    float ps = hv.x * s4.x + hv.y * s4.y + hv.z * s4.z + hv.w * s4.w;
    float pd = hv.x * d4.x + hv.y * d4.y + hv.z * d4.z + hv.w * d4.w;
#pragma unroll
    for (int m = 1; m <= 4; m <<= 1) {
        ps += __shfl_xor(ps, m, 32);
        pd += __shfl_xor(pd, m, 32);
    }
    if ((lane & 7) == 0) {
        int hd = lane >> 3;
        a_src[(size_t)wid * NHEAD + hd] = ps;
        a_dst[(size_t)wid * NHEAD + hd] = pd;
    }
}

__device__ __forceinline__ void edge_pair(const int* __restrict__ ei, int e, int ecnt,
                                          int& s, int& d) {
    if (e < ecnt) { s = ei[e]; d = ei[ecnt + e]; }
    else          { s = d = e - ecnt; }
}

// ---------------- Edge pass 1: segment max (monotonic-uint atomicMax) ----------------
__global__ void edge_max_kernel(const int* __restrict__ ei, const float* __restrict__ a_src,
                                const float* __restrict__ a_dst, unsigned* __restrict__ amax,
                                int ecnt, int n) {
    int t = blockIdx.x * blockDim.x + threadIdx.x;
    if (t >= (ecnt + n) * NHEAD) return;
    int e = t >> 2, hd = t & 3;
    int s, d;
    edge_pair(ei, e, ecnt, s, d);
    float al = a_src[s * NHEAD + hd] + a_dst[d * NHEAD + hd];
    al = al > 0.0f ? al : LRELU * al;
    atomicMax(&amax[d * NHEAD + hd], mapf(al));
}

// ---------------- Edge pass 2: segment sum of exp(alpha - max) ----------------
__global__ void edge_sum_kernel(const int* __restrict__ ei, const float* __restrict__ a_src,
                                const float* __restrict__ a_dst, const unsigned* __restrict__ amax,
                                float* __restrict__ asum, int ecnt, int n) {
    int t = blockIdx.x * blockDim.x + threadIdx.x;
    if (t >= (ecnt + n) * NHEAD) return;
    int e = t >> 2, hd = t & 3;
    int s, d;
    edge_pair(ei, e, ecnt, s, d);
    float al = a_src[s * NHEAD + hd] + a_dst[d * NHEAD + hd];
    al = al > 0.0f ? al : LRELU * al;
    float m = unmapf(amax[d * NHEAD + hd]);
    atomicAdd(&asum[d * NHEAD + hd], expf(al - m));
}

// ---------------- Edge pass 3: weighted gather/scatter, one wave per edge ----------------
__global__ void edge_aggr_kernel(const int* __restrict__ ei, const float* __restrict__ a_src,
                                 const float* __restrict__ a_dst, const unsigned* __restrict__ amax,
                                 const float* __restrict__ asum, const float* __restrict__ Hm,
                                 float* __restrict__ aggr, int ecnt, int n) {
    int wid = (blockIdx.x * blockDim.x + threadIdx.x) >> 5;
    int lane = threadIdx.x & 31;
    if (wid >= ecnt + n) return;
    int s, d;
    edge_pair(ei, wid, ecnt, s, d);
    int hd = lane >> 3;  // lane's float4 (cols 4l..4l+3) lies entirely in head l>>3
    float al = a_src[s * NHEAD + hd] + a_dst[d * NHEAD + hd];
    al = al > 0.0f ? al : LRELU * al;
    float m = unmapf(amax[d * NHEAD + hd]);
    float w = expf(al - m) / (asum[d * NHEAD + hd] + EPS_SM);
    float4 hv = reinterpret_cast<const float4*>(Hm + (size_t)s * DD)[lane];
    float* p = aggr + (size_t)d * DD + lane * 4;
    atomicAdd(p + 0, hv.x * w);
    atomicAdd(p + 1, hv.y * w);
    atomicAdd(p + 2, hv.z * w);
    atomicAdd(p + 3, hv.w * w);
}

// ---------------- Final: residual + bias + LayerNorm2 ----------------
__global__ void final_kernel(const float* __restrict__ xt, const float* __restrict__ aggr,
                             const float* __restrict__ bias, const float* __restrict__ g,
                             const float* __restrict__ b, float* __restrict__ out, int n) {
    int wid = (blockIdx.x * blockDim.x + threadIdx.x) >> 5;
    int lane = threadIdx.x & 31;
    if (wid >= n) return;
    float4 xv = reinterpret_cast<const float4*>(xt + (size_t)wid * DD)[lane];
    float4 av = reinterpret_cast<const float4*>(aggr + (size_t)wid * DD)[lane];
    float4 bi = reinterpret_cast<const float4*>(bias)[lane];
    float4 y;
    y.x = xv.x + av.x + bi.x;
    y.y = xv.y + av.y + bi.y;
    y.z = xv.z + av.z + bi.z;
    y.w = xv.w + av.w + bi.w;
    float mean = waveReduceSum(y.x + y.y + y.z + y.w) * (1.0f / DD);
    float dx = y.x - mean, dy = y.y - mean, dz = y.z - mean, dw = y.w - mean;
    float var = waveReduceSum(dx * dx + dy * dy + dz * dz + dw * dw) * (1.0f / DD);
    float inv = rsqrtf(var + EPS_LN);
    float4 gv = reinterpret_cast<const float4*>(g)[lane];
    float4 bv = reinterpret_cast<const float4*>(b)[lane];
    float4 o;
    o.x = dx * inv * gv.x + bv.x;
    o.y = dy * inv * gv.y + bv.y;
    o.z = dz * inv * gv.z + bv.z;
    o.w = dw * inv * gv.w + bv.w;
    reinterpret_cast<float4*>(out + (size_t)wid * DD)[lane] = o;
}

extern "C" void kernel_launch(void* const* d_in, const int* in_sizes, int n_in,
                              void* d_out, int out_size, void* d_ws, size_t ws_size,
                              hipStream_t stream) {
    const float* x        = (const float*)d_in[0];
    const int*   ei       = (const int*)d_in[1];
    // d_in[2]=edge_attr, d_in[3]=timestamps: unused by the reference math
    const float* W_gat    = (const float*)d_in[4];
    const float* att_src  = (const float*)d_in[5];
    const float* att_dst  = (const float*)d_in[6];
    const float* bias_gat = (const float*)d_in[7];
    const float* ln1_g    = (const float*)d_in[8];
    const float* ln1_b    = (const float*)d_in[9];
    const float* ln2_g    = (const float*)d_in[10];
    const float* ln2_b    = (const float*)d_in[11];

    int n    = in_sizes[0] / DD;
    int ecnt = in_sizes[1] / 2;

    auto align256 = [](size_t v) { return (v + 255) & ~(size_t)255; };
    char* ws = (char*)d_ws;
    size_t nd = (size_t)n * DD * sizeof(float);
    size_t nh = (size_t)n * NHEAD * sizeof(float);
    size_t off = 0;
    float*    xt    = (float*)(ws + off);    off = align256(off + nd);
    float*    hmat  = (float*)(ws + off);    off = align256(off + nd);
    float*    aggr  = (float*)(ws + off);    off = align256(off + nd);
    float*    a_src = (float*)(ws + off);    off = align256(off + nh);
    float*    a_dst = (float*)(ws + off);    off = align256(off + nh);
    unsigned* amax  = (unsigned*)(ws + off); off = align256(off + nh);
    float*    asum  = (float*)(ws + off);    off = align256(off + nh);

    hipMemsetAsync(aggr, 0, nd, stream);
    hipMemsetAsync(amax, 0, nh, stream);   // 0 == monotonic-mapped "-inf" sentinel
    hipMemsetAsync(asum, 0, nh, stream);

    int nodeBlocks = (n + 7) / 8;  // 8 waves/block, one node per wave32
    ln1_kernel<<<nodeBlocks, 256, 0, stream>>>(x, ln1_g, ln1_b, xt, n);
    gemm_wmma_kernel<<<(n + 15) / 16, 256, 0, stream>>>(xt, W_gat, hmat, n);
    score_kernel<<<nodeBlocks, 256, 0, stream>>>(hmat, att_src, att_dst, a_src, a_dst, n);

    int totalEH = (ecnt + n) * NHEAD;
    int ehBlocks = (totalEH + 255) / 256;
    edge_max_kernel<<<ehBlocks, 256, 0, stream>>>(ei, a_src, a_dst, amax, ecnt, n);
    edge_sum_kernel<<<ehBlocks, 256, 0, stream>>>(ei, a_src, a_dst, amax, asum, ecnt, n);

    int edgeWaves = ecnt + n;
    edge_aggr_kernel<<<(edgeWaves + 7) / 8, 256, 0, stream>>>(ei, a_src, a_dst, amax, asum,
                                                              hmat, aggr, ecnt, n);
    final_kernel<<<nodeBlocks, 256, 0, stream>>>(xt, aggr, bias_gat, ln2_g, ln2_b,
                                                 (float*)d_out, n);
}